// SoftSplat_69595650064735
// MI455X (gfx1250) — compile-verified
//
#include <hip/hip_runtime.h>
#include <stdint.h>

#define BDIM 256
#define EPS 1e-7f

// Problem shape is fixed by the reference: x[8,64,256,448].
#define B_DIM 8
#define C_DIM 64
#define H_DIM 256
#define W_DIM 448
#define N_DIM (H_DIM * W_DIM)   // 114688

// ---- TDM availability probing (toolchain differences, see CDNA5_HIP.md) ----
#if __has_include(<hip/amd_detail/amd_gfx1250_TDM.h>)
#define TDM_6ARG 1
#else
#define TDM_6ARG 0
#endif
#define HAVE_TDM __has_builtin(__builtin_amdgcn_tensor_load_to_lds)

typedef uint32_t u32x4 __attribute__((ext_vector_type(4)));
typedef int      i32x4 __attribute__((ext_vector_type(4)));
typedef int      i32x8 __attribute__((ext_vector_type(8)));

__device__ __forceinline__ void atom_add(float* p, float v) {
    // relaxed, agent scope, no return -> global_atomic_add_f32 (STOREcnt path)
    __hip_atomic_fetch_add(p, v, __ATOMIC_RELAXED, __HIP_MEMORY_SCOPE_AGENT);
}

__device__ __forceinline__ int clampi(int v, int lo, int hi) {
    return v < lo ? lo : (v > hi ? hi : v);
}

// One thread = one source pixel. grid = (N/256, B): blockIdx.y = batch,
// blockIdx.x = 256-pixel tile -> no integer division in the prologue, so the
// TDM DMA is issued within the first few instructions.
// x tile (64 channels x 256 pixels, row stride N) is staged into LDS via the
// Tensor Data Mover (two pipelined 32-channel half-tiles). Out-of-bounds
// corners are handled reference-style (clamped index + zero weight) so the
// scatter loop is an unpredicated stream of v_mul + global_atomic_add_f32.
__global__ __launch_bounds__(BDIM) void softsplat_scatter(
    const float* __restrict__ x, const float* __restrict__ flow,
    const float* __restrict__ metric, float* __restrict__ num,
    float* __restrict__ den)
{
    constexpr int C = C_DIM, H = H_DIM, W = W_DIM, N = N_DIM;
    const int tid = threadIdx.x;
    const int b   = blockIdx.y;              // uniform
    const int n0  = blockIdx.x * BDIM;       // uniform
    const int n   = n0 + tid;

    const float* xb = x + (size_t)b * C * N + (size_t)n0;  // uniform tile base
    float* numb     = num + (size_t)b * C * N;

#if HAVE_TDM
    __shared__ float xbuf[C * BDIM];                       // 64 KB tile
    if (tid < 32) {                                        // wave 0 drives the TDM
        const uint32_t lds_base = (uint32_t)(uintptr_t)&xbuf[0];
        #pragma unroll
        for (int h = 0; h < 2; ++h) {                      // two 32-channel half-tiles
            const uint64_t gaddr = (uint64_t)(uintptr_t)(xb + (size_t)h * 32 * N);
            u32x4 g0;
            g0[0] = 1u;                                            // count=1, no gather
            g0[1] = lds_base + (uint32_t)h * (32u * BDIM * 4u);    // lds_addr
            g0[2] = (uint32_t)gaddr;                               // global_addr[31:0]
            g0[3] = (uint32_t)(gaddr >> 32) | (2u << 30);          // global_addr[56:32] | type=2
            i32x8 g1;
            g1[0] = (int)(2u << 16);                               // data_size = 4B; mask=0
            g1[1] = (int)((uint32_t)(BDIM & 0xFFFF) << 16);        // tensor_dim0[15:0] = 256
            g1[2] = (int)(32u << 16);                              // tensor_dim1[15:0] = 32
            g1[3] = (int)((uint32_t)BDIM << 16);                   // tile_dim0 = 256
            g1[4] = 32;                                            // tile_dim1 = 32, tile_dim2 = 0
            g1[5] = (int)(uint32_t)N;                              // tensor_dim0_stride = N
            g1[6] = 0;
            g1[7] = 0;
            i32x4 z4 = {0, 0, 0, 0};
#if TDM_6ARG
            i32x8 z8 = {0, 0, 0, 0, 0, 0, 0, 0};
            __builtin_amdgcn_tensor_load_to_lds(g0, g1, z4, z4, z8, 0);
#else
            __builtin_amdgcn_tensor_load_to_lds(g0, g1, z4, z4, 0);
#endif
        }
    }
#else
    __shared__ float xbuf2[2][BDIM];                       // async double-buffer
    const uint32_t ldsA = (uint32_t)(uintptr_t)&xbuf2[0][tid];
    const uint32_t ldsB = (uint32_t)(uintptr_t)&xbuf2[1][tid];
    uint32_t off = (uint32_t)tid * 4u;                     // byte offset vs xb
    const uint32_t stepN = (uint32_t)N * 4u;
    asm volatile("global_load_async_to_lds_b32 %0, %1, %2"
                 :: "v"(ldsA), "v"(off), "s"(xb) : "memory");
#endif

    // ---- per-pixel flow / weights (overlaps the DMA) ----
    const int iy = n / W;                    // W constexpr -> magic multiply
    const int ix = n - iy * W;
    const size_t bN = (size_t)b * N;
    const float fxl = flow[2 * bN + (size_t)n];
    const float fyl = flow[2 * bN + (size_t)N + (size_t)n];
    const float m   = expf(metric[bN + (size_t)n]);

    const float tx = (float)ix + fxl;
    const float ty = (float)iy + fyl;
    const float x0f = floorf(tx), y0f = floorf(ty);
    const int x0 = (int)x0f, y0 = (int)y0f;
    const int x1 = x0 + 1, y1 = y0 + 1;
    const float fx = tx - x0f, fy = ty - y0f;
    const bool vx0 = (x0 >= 0) & (x0 < W);
    const bool vx1 = (x1 >= 0) & (x1 < W);
    const bool vy0 = (y0 >= 0) & (y0 < H);
    const bool vy1 = (y1 >= 0) & (y1 < H);
    // m-scaled corner weights, zeroed when the corner is out of bounds
    const float a00 = (vx0 & vy0) ? m * (1.f - fx) * (1.f - fy) : 0.f;
    const float a10 = (vx1 & vy0) ? m * fx * (1.f - fy)         : 0.f;
    const float a01 = (vx0 & vy1) ? m * (1.f - fx) * fy         : 0.f;
    const float a11 = (vx1 & vy1) ? m * fx * fy                 : 0.f;
    // clamped (always in-bounds) corner indices, reference-style
    const int cx0 = clampi(x0, 0, W - 1), cx1 = clampi(x1, 0, W - 1);
    const int cy0 = clampi(y0, 0, H - 1), cy1 = clampi(y1, 0, H - 1);
    const int i00 = cy0 * W + cx0, i10 = cy0 * W + cx1;
    const int i01 = cy1 * W + cx0, i11 = cy1 * W + cx1;

    // denominator scatter (4 unpredicated atomics)
    float* denb = den + bN;
    atom_add(denb + i00, a00);
    atom_add(denb + i10, a10);
    atom_add(denb + i01, a01);
    atom_add(denb + i11, a11);

#if HAVE_TDM
    if (tid < 32) __builtin_amdgcn_s_wait_tensorcnt(1);    // half-tile 0 landed
    __syncthreads();
    #pragma unroll 8
    for (int c = 0; c < 32; ++c) {
        const float xv = xbuf[c * BDIM + tid];
        float* pc = numb + (size_t)c * N;
        atom_add(pc + i00, xv * a00);
        atom_add(pc + i10, xv * a10);
        atom_add(pc + i01, xv * a01);
        atom_add(pc + i11, xv * a11);
    }
    if (tid < 32) __builtin_amdgcn_s_wait_tensorcnt(0);    // half-tile 1 landed
    __syncthreads();
    #pragma unroll 8
    for (int c = 32; c < C; ++c) {
        const float xv = xbuf[c * BDIM + tid];
        float* pc = numb + (size_t)c * N;
        atom_add(pc + i00, xv * a00);
        atom_add(pc + i10, xv * a10);
        atom_add(pc + i01, xv * a01);
        atom_add(pc + i11, xv * a11);
    }
#else
    #pragma unroll 4
    for (int c = 0; c < C; ++c) {
        off += stepN;
        if (c + 1 < C) {
            asm volatile("global_load_async_to_lds_b32 %0, %1, %2"
                         :: "v"((c & 1) ? ldsA : ldsB), "v"(off), "s"(xb) : "memory");
            asm volatile("s_wait_asynccnt 1" ::: "memory");
        } else {
            asm volatile("s_wait_asynccnt 0" ::: "memory");
        }
        const float xv = xbuf2[c & 1][tid];
        float* pc = numb + (size_t)c * N;
        atom_add(pc + i00, xv * a00);
        atom_add(pc + i10, xv * a10);
        atom_add(pc + i01, xv * a01);
        atom_add(pc + i11, xv * a11);
    }
#endif
}

// out[b,c,n] = num[b,c,n] / (den[b,n] + EPS), 4 pixels per thread (float4).
// grid = (C*N/4/256, B): all index math is against constexpr dims.
__global__ __launch_bounds__(BDIM) void softsplat_normalize(
    float* __restrict__ out, const float* __restrict__ den)
{
    constexpr int C = C_DIM, N = N_DIM;
    constexpr int N4 = N / 4;
    const int b   = blockIdx.y;
    const int idx = blockIdx.x * BDIM + threadIdx.x;   // float4 index within batch
    const int n4  = idx % N4;                          // constexpr -> magic multiply

    float4* outb = (float4*)(out + (size_t)b * C * N);
    float4 v = outb[idx];
    const float4 d = *(const float4*)(den + (size_t)b * N + (size_t)n4 * 4);
    v.x /= (d.x + EPS);
    v.y /= (d.y + EPS);
    v.z /= (d.z + EPS);
    v.w /= (d.w + EPS);
    outb[idx] = v;
}

extern "C" void kernel_launch(void* const* d_in, const int* in_sizes, int n_in,
                              void* d_out, int out_size, void* d_ws, size_t ws_size,
                              hipStream_t stream) {
    const float* x      = (const float*)d_in[0];   // [8,64,256,448]
    const float* flow   = (const float*)d_in[1];   // [8,2,256,448]
    const float* metric = (const float*)d_in[2];   // [8,1,256,448]
    float* out = (float*)d_out;                    // [8,64,256,448] numerator -> result
    float* den = (float*)d_ws;                     // [8,256,448] denominator (3.67 MB)

    hipMemsetAsync(d_out, 0, (size_t)out_size * sizeof(float), stream);
    hipMemsetAsync(d_ws, 0, (size_t)B_DIM * N_DIM * sizeof(float), stream);

    dim3 sgrid(N_DIM / BDIM, B_DIM);               // (448, 8)
    softsplat_scatter<<<sgrid, BDIM, 0, stream>>>(x, flow, metric, out, den);

    dim3 ngrid((C_DIM * N_DIM / 4) / BDIM, B_DIM); // (7168, 8), exact
    softsplat_normalize<<<ngrid, BDIM, 0, stream>>>(out, den);
}